// RawRoutedMoA_8022998909801
// MI455X (gfx1250) — compile-verified
//
#include <hip/hip_runtime.h>
#include <math.h>

typedef __attribute__((ext_vector_type(2))) float v2f;
typedef __attribute__((ext_vector_type(8))) float v8f;
typedef __attribute__((ext_vector_type(4))) unsigned int v4u;
typedef __attribute__((ext_vector_type(8))) int v8i;
typedef __attribute__((ext_vector_type(4))) int v4i;

#define BB 256
#define TT 512
#define DD 512
#define NE 5
#define HID 64
#define DOUTN 128

// workspace layout (floats)
#define WS_GATES  0                       // B*5
#define WS_S      1280                    // B*T attn logits
#define WS_A      (WS_S    + BB*TT)       // B*T attn weights
#define WS_MEAN   (WS_A    + BB*TT)       // B*D
#define WS_MAX    (WS_MEAN + BB*DD)       // B*D
#define WS_ATP    (WS_MAX  + BB*DD)       // B*D
#define WS_LAST   (WS_ATP  + BB*DD)       // B*D
#define WS_ZPART  (WS_LAST + BB*DD)       // B*128 (2 halves x 64 oc)
#define WS_WT     (WS_ZPART+ BB*128)      // 8*512*64 repacked conv weights

__device__ __forceinline__ float gelu_exact(float x) {
    return 0.5f * x * (1.0f + erff(x * 0.70710678118654752f));
}

// ---- repack conv weights w[oc,c,k] -> Wt[k][c][oc] (coalesced B-frag loads) ----
__global__ void repack_kernel(const float* __restrict__ w, float* __restrict__ Wt) {
    int idx = blockIdx.x * blockDim.x + threadIdx.x;   // 64*512*8 = 262144
    int oc  = idx >> 12;
    int rem = idx & 4095;
    int c   = rem >> 3;
    int k   = rem & 7;
    Wt[(k * 512 + c) * 64 + oc] = w[idx];
}

// ---- router: conv16x32 -> gelu -> group mean -> 64 -> 5 logits -> top2 softmax ----
__global__ void router_kernel(const float* __restrict__ raw,
                              const float* __restrict__ cw, const float* __restrict__ cb,
                              const float* __restrict__ hw, const float* __restrict__ hb,
                              float* __restrict__ gates) {
    __shared__ float f[64];
    __shared__ float lg[5];
    int b = blockIdx.x, tid = threadIdx.x;   // 64 threads
    const float* x = raw + b * 512;
    {
        int oc = tid >> 2, g = tid & 3;
        float acc = 0.0f;
        for (int pi = 0; pi < 8; ++pi) {
            int p = g * 8 + pi;
            float cacc = cb[oc];
            for (int k = 0; k < 32; ++k) {
                int ii = 16 * p - 8 + k;
                float xv = (ii >= 0 && ii < 512) ? x[ii] : 0.0f;
                cacc += xv * cw[oc * 32 + k];
            }
            acc += gelu_exact(cacc);
        }
        f[tid] = acc * 0.125f;
    }
    __syncthreads();
    if (tid < 5) {
        float l = hb[tid];
        for (int j = 0; j < 64; ++j) l += f[j] * hw[j * 5 + tid];
        lg[tid] = l;
    }
    __syncthreads();
    if (tid == 0) {
        int i1 = 0; float v1 = lg[0];
        for (int e = 1; e < 5; ++e) if (lg[e] > v1) { v1 = lg[e]; i1 = e; }
        int i2 = -1; float v2 = -INFINITY;
        for (int e = 0; e < 5; ++e) { if (e == i1) continue; if (lg[e] > v2) { v2 = lg[e]; i2 = e; } }
        float t  = expf(v2 - v1);
        float w1 = 1.0f / (1.0f + t);
        float w2 = t / (1.0f + t);
        for (int e = 0; e < 5; ++e) gates[b * 5 + e] = 0.0f;
        gates[b * 5 + i1] = w1;
        gates[b * 5 + i2] = w2;
    }
}

// ---- attn logits: s[b,t] = h[b,t,:].aw + ab  (one wave per row, float4 loads) ----
__global__ void attn_logits_kernel(const float* __restrict__ h,
                                   const float* __restrict__ aw, const float* __restrict__ ab,
                                   const float* __restrict__ gates, float* __restrict__ s) {
    int b = blockIdx.x >> 6;                 // 64 blocks per b (8 rows per block)
    if (gates[b * NE + 3] == 0.0f) return;
    int lane = threadIdx.x & 31;
    int wid  = threadIdx.x >> 5;
    int r = blockIdx.x * 8 + wid;            // global row = b*T + t
    const float4* hp4 = reinterpret_cast<const float4*>(h + (size_t)r * DD);
    const float4* aw4 = reinterpret_cast<const float4*>(aw);
    float p = 0.0f;
#pragma unroll
    for (int i = 0; i < 4; ++i) {
        float4 hv = hp4[i * 32 + lane];
        float4 wv = aw4[i * 32 + lane];
        p += hv.x * wv.x + hv.y * wv.y + hv.z * wv.z + hv.w * wv.w;
    }
#pragma unroll
    for (int off = 16; off; off >>= 1) p += __shfl_xor(p, off, 32);
    if (lane == 0) s[r] = p + ab[0];
}

__global__ void attn_softmax_kernel(const float* __restrict__ s, float* __restrict__ a) {
    __shared__ float red[512];
    int b = blockIdx.x, tid = threadIdx.x;
    float v = s[b * TT + tid];
    red[tid] = v;
    for (int off = 256; off; off >>= 1) { __syncthreads(); if (tid < off) red[tid] = fmaxf(red[tid], red[tid + off]); }
    __syncthreads();
    float m = red[0];
    __syncthreads();
    float e = expf(v - m);
    red[tid] = e;
    for (int off = 256; off; off >>= 1) { __syncthreads(); if (tid < off) red[tid] += red[tid + off]; }
    __syncthreads();
    a[b * TT + tid] = e / red[0];
}

// ---- single pass over h: mean/max/last/attn-pooled per (b,d) ----
__global__ void pool_kernel(const float* __restrict__ h, const float* __restrict__ a,
                            float* __restrict__ mean, float* __restrict__ mx,
                            float* __restrict__ atp, float* __restrict__ last) {
    int b = blockIdx.x, d = threadIdx.x;     // 512 threads
    const float* hb = h + (size_t)b * TT * DD;
    const float* ab = a + b * TT;
    float s = 0.0f, m = -INFINITY, w = 0.0f;
    for (int t = 0; t < TT; ++t) {
        float v = hb[t * DD + d];
        s += v;
        m = fmaxf(m, v);
        w += ab[t] * v;
    }
    mean[b * DD + d] = s * (1.0f / TT);
    mx[b * DD + d]   = m;
    atp[b * DD + d]  = w;
    last[b * DD + d] = hb[(TT - 1) * DD + d];
}

// ---- experts 0..3: gated MLPs, write gated sum to out ----
__global__ void mlp_kernel(const float* __restrict__ gates,
                           const float* __restrict__ mean, const float* __restrict__ last,
                           const float* __restrict__ mx,   const float* __restrict__ atp,
                           const float* mp_w1, const float* mp_b1, const float* mp_w2, const float* mp_b2,
                           const float* lt_w1, const float* lt_b1, const float* lt_w2, const float* lt_b2,
                           const float* mx_w1, const float* mx_b1, const float* mx_w2, const float* mx_b2,
                           const float* ap_w1, const float* ap_b1, const float* ap_w2, const float* ap_b2,
                           float* __restrict__ out) {
    __shared__ float hid[64];
    int b = blockIdx.x, tid = threadIdx.x;   // 128 threads
    const float* xs[4]  = { mean + b * DD, last + b * DD, mx + b * DD, atp + b * DD };
    const float* w1s[4] = { mp_w1, lt_w1, mx_w1, ap_w1 };
    const float* b1s[4] = { mp_b1, lt_b1, mx_b1, ap_b1 };
    const float* w2s[4] = { mp_w2, lt_w2, mx_w2, ap_w2 };
    const float* b2s[4] = { mp_b2, lt_b2, mx_b2, ap_b2 };
    float acc = 0.0f;
    for (int e = 0; e < 4; ++e) {
        float g = gates[b * NE + e];
        if (g == 0.0f) continue;             // block-uniform
        if (tid < 64) {
            float hsum = b1s[e][tid];
            const float* x  = xs[e];
            const float* w1 = w1s[e];
            for (int c = 0; c < DD; ++c) hsum += x[c] * w1[c * 64 + tid];
            hid[tid] = gelu_exact(hsum);
        }
        __syncthreads();
        float o = b2s[e][tid];
        const float* w2 = w2s[e];
        for (int hh = 0; hh < 64; ++hh) o += hid[hh] * w2[hh * DOUTN + tid];
        acc += g * o;
        __syncthreads();
    }
    out[b * DOUTN + tid] = acc;
}

// ---- expert 4 conv as WMMA f32 GEMM: M=128 ops, N=64 oc, K=4096 (c,k) ----
// block per b, 256 threads = 8 waves = (4 oc-tiles) x (2 op-halves); each wave: 4 M-tiles.
// h[b] staged through LDS in 32-channel slices by the Tensor Data Mover; the TDM pad
// feature (1 dword pad every 32 dwords) produces the pitch-33 conflict-free layout.
#define CCHUNK 32
#define PITCH  33
#define LDSBYTES (516 * PITCH * 4)
__global__ void conv_wmma_kernel(const float* __restrict__ h,
                                 const float* __restrict__ Wt,
                                 const float* __restrict__ convb,
                                 const float* __restrict__ gates,
                                 float* __restrict__ zpart) {
    extern __shared__ float sh[];
    const int b = blockIdx.x;
    if (gates[b * NE + 4] == 0.0f) return;   // block-uniform skip
    const int tid  = threadIdx.x;
    const int lane = tid & 31;
    const int wid  = tid >> 5;
    const int oc0  = (wid & 3) * 16;
    const int mh   = wid >> 2;               // op half: 0 -> ops 0..63, 1 -> ops 64..127
    const int ln16 = lane & 15;
    const int kkb  = (lane < 16) ? 0 : 2;    // K-pair base per lane half

    // zero pad rows: t = -2,-1 (rows 0,1) and t = 512,513 (rows 514,515)
    if (tid < 128) {
        int rsel = tid >> 5;
        int r = (rsel < 2) ? rsel : (rsel + 512);
        sh[r * PITCH + (tid & 31)] = 0.0f;
    }

    v8f acc[4] = {};
    const float* hb = h + (size_t)b * TT * DD;
    // LDS byte address of row 2 (t=0) relative to the wave's LDS window
    const unsigned lds_dst = __builtin_amdgcn_groupstaticsize() + 2u * PITCH * 4u;

    for (int cc = 0; cc < DD / CCHUNK; ++cc) {
        __syncthreads();                     // previous chunk fully consumed
        if (wid == 0) {
            // TDM descriptor: 512x32-float tile, row stride 512 floats, dst LDS with
            // pad_enable: 1 dword pad after every 32 dwords -> pitch 33.
            unsigned long long ga = (unsigned long long)(const void*)(hb + cc * CCHUNK);
            v4u g0;
            g0.x = 1u;                                               // count=1, user descriptor
            g0.y = lds_dst;                                          // lds_addr (bytes)
            g0.z = (unsigned)ga;                                     // global_addr[31:0]
            g0.w = (unsigned)((ga >> 32) & 0x01FFFFFFu) | (2u << 30);// global_addr[56:32] | type=2
            v8i g1;
            g1[0] = (2 << 16) | (1 << 20) | (4 << 22);  // data_size=4B, pad_enable, interval=32dw, amount=1dw
            g1[1] = (int)(32u << 16);                   // tensor_dim0 = 32
            g1[2] = (int)(512u << 16);                  // tensor_dim1 = 512 (lo16)
            g1[3] = (int)(32u << 16);                   // tile_dim0 = 32
            g1[4] = 512;                                // tile_dim1 = 512, tile_dim2 = 0
            g1[5] = 512;                                // tensor_dim0_stride = 512 (lo32)
            g1[6] = 0;                                  // stride hi / dim1_stride lo
            g1[7] = 0;
            v4i z4 = {0, 0, 0, 0};
            v8i z8 = {0, 0, 0, 0, 0, 0, 0, 0};
            __builtin_amdgcn_tensor_load_to_lds(g0, g1, z4, z4, z8, 0);
            __builtin_amdgcn_s_wait_tensorcnt(0);
        }
        __syncthreads();                     // publish DMA'd chunk to all waves
        for (int k = 0; k < 8; ++k) {
#pragma unroll
            for (int c4 = 0; c4 < CCHUNK / 4; ++c4) {
                int cg = cc * CCHUNK + c4 * 4;
                const float* wp = Wt + (size_t)(k * 512 + cg + kkb) * 64 + oc0 + ln16;
                v2f bfrag;
                bfrag.x = wp[0];     // B[K=kkb,   N]
                bfrag.y = wp[64];    // B[K=kkb+1, N]
#pragma unroll
                for (int mt = 0; mt < 4; ++mt) {
                    int op  = mh * 64 + mt * 16 + ln16;
                    int row = 4 * op + k;            // (t = 4*op - 2 + k) + 2
                    const float* ap = &sh[row * PITCH + c4 * 4 + kkb];
                    v2f afrag;
                    afrag.x = ap[0];                 // A[m, K=kkb]
                    afrag.y = ap[1];                 // A[m, K=kkb+1]
                    acc[mt] = __builtin_amdgcn_wmma_f32_16x16x4_f32(
                        false, afrag, false, bfrag, (short)0, acc[mt], false, false);
                }
            }
        }
    }

    // bias + gelu + sum over the 64 ops this wave owns
    float bias = convb[oc0 + ln16];
    float s = 0.0f;
#pragma unroll
    for (int mt = 0; mt < 4; ++mt)
#pragma unroll
        for (int r = 0; r < 8; ++r)
            s += gelu_exact(acc[mt][r] + bias);
    s += __shfl_xor(s, 16, 32);              // combine M halves (lanes l / l+16)
    if (lane < 16) zpart[b * 128 + mh * 64 + oc0 + lane] = s;   // disjoint slots: deterministic
}

// ---- expert 4 head: out += gate4 * ((zsum/128) @ cv_out_w + cv_out_b) ----
__global__ void finalize_kernel(const float* __restrict__ gates, const float* __restrict__ zpart,
                                const float* __restrict__ cvw, const float* __restrict__ cvb,
                                float* __restrict__ out) {
    int b = blockIdx.x, j = threadIdx.x;     // 128 threads
    float g = gates[b * NE + 4];
    if (g == 0.0f) return;
    const float* zp = zpart + b * 128;
    float o = cvb[j];
    for (int hh = 0; hh < 64; ++hh)
        o += ((zp[hh] + zp[64 + hh]) * (1.0f / 128.0f)) * cvw[hh * DOUTN + j];
    out[b * DOUTN + j] += g * o;
}

extern "C" void kernel_launch(void* const* d_in, const int* in_sizes, int n_in,
                              void* d_out, int out_size, void* d_ws, size_t ws_size,
                              hipStream_t stream) {
    (void)in_sizes; (void)n_in; (void)out_size; (void)ws_size;
    const float* h    = (const float*)d_in[0];
    const float* raw  = (const float*)d_in[1];
    const float* rcw  = (const float*)d_in[2];
    const float* rcb  = (const float*)d_in[3];
    const float* rhw  = (const float*)d_in[4];
    const float* rhb  = (const float*)d_in[5];
    const float* mp_w1=(const float*)d_in[6],  *mp_b1=(const float*)d_in[7];
    const float* mp_w2=(const float*)d_in[8],  *mp_b2=(const float*)d_in[9];
    const float* lt_w1=(const float*)d_in[10], *lt_b1=(const float*)d_in[11];
    const float* lt_w2=(const float*)d_in[12], *lt_b2=(const float*)d_in[13];
    const float* mx_w1=(const float*)d_in[14], *mx_b1=(const float*)d_in[15];
    const float* mx_w2=(const float*)d_in[16], *mx_b2=(const float*)d_in[17];
    const float* apw  =(const float*)d_in[18], *apb  =(const float*)d_in[19];
    const float* ap_w1=(const float*)d_in[20], *ap_b1=(const float*)d_in[21];
    const float* ap_w2=(const float*)d_in[22], *ap_b2=(const float*)d_in[23];
    const float* cvw  =(const float*)d_in[24], *cvb  =(const float*)d_in[25];
    const float* cow  =(const float*)d_in[26], *cob  =(const float*)d_in[27];
    float* out = (float*)d_out;
    float* W   = (float*)d_ws;
    float* gates = W + WS_GATES;
    float* s     = W + WS_S;
    float* a     = W + WS_A;
    float* mean  = W + WS_MEAN;
    float* mx    = W + WS_MAX;
    float* atp   = W + WS_ATP;
    float* last  = W + WS_LAST;
    float* zpart = W + WS_ZPART;
    float* Wt    = W + WS_WT;

    repack_kernel      <<<512, 512, 0, stream>>>(cvw, Wt);
    router_kernel      <<<BB, 64, 0, stream>>>(raw, rcw, rcb, rhw, rhb, gates);
    attn_logits_kernel <<<(BB * TT) / 8, 256, 0, stream>>>(h, apw, apb, gates, s);
    attn_softmax_kernel<<<BB, 512, 0, stream>>>(s, a);
    pool_kernel        <<<BB, 512, 0, stream>>>(h, a, mean, mx, atp, last);
    mlp_kernel         <<<BB, 128, 0, stream>>>(gates, mean, last, mx, atp,
                                                mp_w1, mp_b1, mp_w2, mp_b2,
                                                lt_w1, lt_b1, lt_w2, lt_b2,
                                                mx_w1, mx_b1, mx_w2, mx_b2,
                                                ap_w1, ap_b1, ap_w2, ap_b2, out);
    conv_wmma_kernel   <<<BB, 256, LDSBYTES, stream>>>(h, Wt, cvb, gates, zpart);
    finalize_kernel    <<<BB, 128, 0, stream>>>(gates, zpart, cow, cob, out);
}